// RNN_5815385718926
// MI455X (gfx1250) — compile-verified
//
#include <hip/hip_runtime.h>
#include <math.h>

// Problem dims (from reference)
#define BATCH    512
#define TIME     28
#define SIZE_IN  28
#define SIZE_RES 2000
#define SIZE_OUT 10
#define NSTEP    (BATCH * TIME)      // 14336 sequential scan steps

typedef __attribute__((ext_vector_type(2))) float v2f;
typedef __attribute__((ext_vector_type(8))) float v8f;
typedef __attribute__((ext_vector_type(4))) unsigned int u32x4;
typedef __attribute__((ext_vector_type(4))) int i32x4v;
typedef __attribute__((ext_vector_type(8))) int i32x8v;

// ---------------------------------------------------------------------------
// Kernel 0: init hidden-state double buffer + grid-barrier counter
// ---------------------------------------------------------------------------
__global__ void rnn_init_kernel(float* hbuf, unsigned* cnt) {
    int i = blockIdx.x * blockDim.x + threadIdx.x;
    if (i < 2 * 2048) hbuf[i] = 0.0f;
    if (i == 0) *cnt = 0u;
}

// ---------------------------------------------------------------------------
// Kernel A: U(14336x2000) = X(14336x28) @ W_in^T  via V_WMMA_F32_16X16X4_F32
// ---------------------------------------------------------------------------
#define MT (NSTEP / 16)     // 896 row tiles
#define NT (SIZE_RES / 16)  // 125 col tiles

__global__ void u_gemm_wmma(const float* __restrict__ X,
                            const float* __restrict__ Win,
                            float* __restrict__ U) {
    int wave = (blockIdx.x * blockDim.x + threadIdx.x) >> 5;
    int lane = threadIdx.x & 31;
    if (wave >= MT * NT) return;
    int tm   = wave / NT;
    int tn   = wave % NT;
    int ln   = lane & 15;       // M index for A, N index for B
    int half = lane >> 4;       // selects K pair
    int kb   = half * 2;

    const float* arow = X   + (size_t)(tm * 16 + ln) * SIZE_IN;
    const float* brow = Win + (size_t)(tn * 16 + ln) * SIZE_IN;

    v8f c = {};
#pragma unroll
    for (int k0 = 0; k0 < SIZE_IN; k0 += 4) {
        v2f a = *(const v2f*)(arow + k0 + kb);
        v2f b = *(const v2f*)(brow + k0 + kb);
        c = __builtin_amdgcn_wmma_f32_16x16x4_f32(false, a, false, b,
                                                  (short)0, c, false, false);
    }
    float* dcol = U + (size_t)tn * 16 + ln;
#pragma unroll
    for (int v = 0; v < 8; ++v) {
        int rm = v + 8 * half;
        dcol[(size_t)(tm * 16 + rm) * SIZE_RES] = c[v];
    }
}

// ---------------------------------------------------------------------------
// Kernel B: persistent-grid sequential scan.
//   63 WGs x 256 threads, 32 rows of W_res per WG held resident in LDS for
//   all 14336 steps; slice staged via Tensor Data Mover (tensor_load_to_lds).
// ---------------------------------------------------------------------------
#define RPW 32
#define GWG 63   // 63*32 = 2016 >= 2000 rows

__global__ void __launch_bounds__(256, 1)
rnn_recurrence(const float* __restrict__ Wres,
               const float* __restrict__ U,
               float* hbuf,                   // 2 * 2048 floats (double buffer)
               float* __restrict__ Hlast,     // 512 * 2000
               unsigned* cnt) {
    extern __shared__ float lds[];
    float* Wl = lds;                       // RPW * SIZE_RES floats
    float* hl = lds + RPW * SIZE_RES;      // 2048 floats (h broadcast copy)

    const int tid   = threadIdx.x;
    const int r0    = blockIdx.x * RPW;
    const int nrows = (SIZE_RES - r0 < RPW) ? (SIZE_RES - r0) : RPW;

    // ---- Stage this WG's W_res slice into LDS with the TDM --------------
    // 2-D descriptor: tile = nrows x 2000 f32, row stride 2000 elements.
    if (tid < 32) {   // wave 0 only: TDM op ignores EXEC, issue once per WG
        unsigned long long ga =
            (unsigned long long)(const void*)(Wres + (size_t)r0 * SIZE_RES);
        unsigned lds_base = __builtin_amdgcn_groupstaticsize(); // dynamic LDS start

        u32x4 g0;
        g0[0] = 1u;                                   // count=1, user mode, no gather
        g0[1] = lds_base;                             // lds_addr (bytes)
        g0[2] = (unsigned)(ga & 0xFFFFFFFFull);       // global_addr[31:0]
        g0[3] = (unsigned)((ga >> 32) & 0x01FFFFFFull)// global_addr[56:32]
              | (2u << 30);                           // type=2 ("image")

        i32x8v g1;
        g1[0] = (int)(2u << 16);                      // data_size=2 (4 bytes)
        g1[1] = (int)(2000u << 16);                   // tensor_dim0 = 2000 (lo16)
        g1[2] = (int)((unsigned)nrows << 16);         // tensor_dim1 = nrows (lo16)
        g1[3] = (int)(2000u << 16);                   // tile_dim0 = 2000
        g1[4] = nrows;                                // tile_dim1 = nrows, tile_dim2 = 0
        g1[5] = 2000;                                 // tensor_dim0_stride[31:0]
        g1[6] = 0;                                    // stride hi / dim1_stride (unused, 2D)
        g1[7] = 0;

        i32x4v gz = {0, 0, 0, 0};                     // groups 2/3: unused (2-D)
#if defined(__has_include) && __has_include(<hip/amd_detail/amd_gfx1250_TDM.h>)
        i32x8v gz8 = {0, 0, 0, 0, 0, 0, 0, 0};
        __builtin_amdgcn_tensor_load_to_lds(g0, g1, gz, gz, gz8, 0);
#else
        __builtin_amdgcn_tensor_load_to_lds(g0, g1, gz, gz, 0);
#endif
        __builtin_amdgcn_s_wait_tensorcnt(0);
    }
    __syncthreads();

    const int row = tid >> 3;   // 0..31
    const int s8  = tid & 7;    // 8 threads cooperate per row
    const float2* w2 = (const float2*)(Wl + row * SIZE_RES);  // 8B aligned (8000B rows)
    const float2* h2 = (const float2*)hl;

    for (int t = 0; t < NSTEP; ++t) {
        const float* hg = hbuf + (t & 1) * 2048;
        float*       hn = hbuf + ((t + 1) & 1) * 2048;

        // Broadcast h into LDS with device-scope 8B (acquire-side) loads
        {
            const unsigned long long* hg64 = (const unsigned long long*)hg;
            unsigned long long*       hl64 = (unsigned long long*)hl;
            for (int i = tid; i < SIZE_RES / 2; i += 256)
                hl64[i] = __hip_atomic_load(hg64 + i, __ATOMIC_RELAXED,
                                            __HIP_MEMORY_SCOPE_AGENT);
        }
        if (tid == 0 && t + 1 < NSTEP)
            __builtin_prefetch(U + (size_t)(t + 1) * SIZE_RES + r0, 0, 1);
        __syncthreads();

        // Partial dot product, b64 LDS reads: float2 index j = s8 + 8i
        // -> word addrs {row*16 + 2*s8, +1}: 4 rows x 8 sublanes cover all 64 banks
        float acc = 0.0f;
        if (row < nrows) {
#pragma unroll 5
            for (int i = 0; i < SIZE_RES / 16; ++i) {   // 125 iterations
                int j = s8 + 8 * i;
                float2 w = w2[j];
                float2 h = h2[j];
                acc = fmaf(w.x, h.x, acc);
                acc = fmaf(w.y, h.y, acc);
            }
        }
        // width-8 intra-wave reduction
        acc += __shfl_down(acc, 4, 8);
        acc += __shfl_down(acc, 2, 8);
        acc += __shfl_down(acc, 1, 8);

        if (s8 == 0 && row < nrows) {
            int r = r0 + row;
            float hv = tanhf(U[(size_t)t * SIZE_RES + r] + acc);
            __hip_atomic_store(hn + r, hv, __ATOMIC_RELAXED,
                               __HIP_MEMORY_SCOPE_AGENT);
            if ((t % TIME) == (TIME - 1))
                Hlast[(size_t)(t / TIME) * SIZE_RES + r] = hv;
        }
        __syncthreads();

        // Grid barrier: monotone counter, target = (t+1)*GWG (max 903168 < 2^32)
        if (tid == 0) {
            __threadfence();
            atomicAdd(cnt, 1u);
            unsigned target = (unsigned)(t + 1) * (unsigned)GWG;
            while (atomicAdd(cnt, 0u) < target) __builtin_amdgcn_s_sleep(1);
            __threadfence();
        }
        __syncthreads();
    }
}

// ---------------------------------------------------------------------------
// Kernel C: out(512x10) = Hlast(512x2000) @ W_out^T  via WMMA f32 16x16x4.
// ---------------------------------------------------------------------------
__global__ void out_gemm_wmma(const float* __restrict__ Hlast,
                              const float* __restrict__ Wout,
                              float* __restrict__ out) {
    int wave = (blockIdx.x * blockDim.x + threadIdx.x) >> 5;
    int lane = threadIdx.x & 31;
    if (wave >= BATCH / 16) return;   // 32 tiles
    int tm   = wave;
    int ln   = lane & 15;
    int half = lane >> 4;
    int kb   = half * 2;

    const float* arow = Hlast + (size_t)(tm * 16 + ln) * SIZE_RES;
    const bool   bval = (ln < SIZE_OUT);
    const float* brow = Wout + (size_t)(bval ? ln : 0) * SIZE_RES;
    v2f bz = {0.0f, 0.0f};

    v8f c = {};
#pragma unroll 4
    for (int k0 = 0; k0 < SIZE_RES; k0 += 4) {
        v2f a = *(const v2f*)(arow + k0 + kb);
        v2f b = bval ? *(const v2f*)(brow + k0 + kb) : bz;
        c = __builtin_amdgcn_wmma_f32_16x16x4_f32(false, a, false, b,
                                                  (short)0, c, false, false);
    }
    if (bval) {
#pragma unroll
        for (int v = 0; v < 8; ++v) {
            int rm = v + 8 * half;
            out[(size_t)(tm * 16 + rm) * SIZE_OUT + ln] = c[v];
        }
    }
}

// ---------------------------------------------------------------------------
// Launch: init -> input GEMM (WMMA) -> persistent scan (TDM) -> output GEMM
// Workspace (floats): U 28,672,000 | hbuf 4,096 | Hlast 1,024,000 | cnt
// ---------------------------------------------------------------------------
extern "C" void kernel_launch(void* const* d_in, const int* in_sizes, int n_in,
                              void* d_out, int out_size, void* d_ws, size_t ws_size,
                              hipStream_t stream) {
    const float* X    = (const float*)d_in[0];
    const float* Win  = (const float*)d_in[1];
    const float* Wres = (const float*)d_in[2];
    const float* Wout = (const float*)d_in[3];
    float* out = (float*)d_out;

    float* ws    = (float*)d_ws;
    float* U     = ws;
    float* hbuf  = U + (size_t)NSTEP * SIZE_RES;
    float* Hlast = hbuf + 2 * 2048;
    unsigned* cnt = (unsigned*)(Hlast + (size_t)BATCH * SIZE_RES);

    rnn_init_kernel<<<(2 * 2048 + 255) / 256, 256, 0, stream>>>(hbuf, cnt);

    int wavesA  = MT * NT;                  // 112,000 tiles
    int blocksA = (wavesA + 7) / 8;         // 8 waves (256 thr) per block
    u_gemm_wmma<<<blocksA, 256, 0, stream>>>(X, Win, U);

    size_t ldsB = (size_t)(RPW * SIZE_RES + 2048) * sizeof(float);  // 264,192 B
    rnn_recurrence<<<GWG, 256, ldsB, stream>>>(Wres, U, hbuf, Hlast, cnt);

    out_gemm_wmma<<<(BATCH / 16 + 7) / 8, 256, 0, stream>>>(Hlast, Wout, out);
}